// LoopyBeliefPropagation_72146860638850
// MI455X (gfx1250) — compile-verified
//
#include <hip/hip_runtime.h>
#include <hip/hip_bf16.h>
#include <math.h>

// CDNA5 / gfx1250, wave32.
typedef __attribute__((ext_vector_type(2))) float v2f;
typedef __attribute__((ext_vector_type(8))) float v8f;

#define LDIM 128
#define MS   130   // padded LDS row stride for M (kills 64-bank conflicts)
#define CS   129   // padded LDS row stride for C1
#define LN2F 0.6931471805599453f

__device__ __forceinline__ float softplusf(float x) {
  // log(1 + exp(x)), stable
  return fmaxf(x, 0.f) + log1pf(expf(-fabsf(x)));
}

__global__ __launch_bounds__(256)
void LoopyBeliefPropagation_72146860638850_kernel(
    const float* __restrict__ s_edge,          // [B,L,L,2]
    const float* __restrict__ s_sib,           // [B,L,L,L]
    const unsigned char* __restrict__ mask,    // [B,L,L] bool
    float* __restrict__ out)                   // [B,L,L,2]
{
  __shared__ float M[2][LDIM][MS];    // message plane m_sib[h, d, s, b, v] for this (b,h)
  __shared__ float C1[LDIM][CS];      // softplus(s_sib[b, s, h, d])
  __shared__ float pe[LDIM][2];       // p_edge[d,v] = s_edge[b, d, h, v]
  __shared__ float yv[2][LDIM];       // masked row sums
  __shared__ float bv[2][LDIM];       // beliefs
  __shared__ float mrow[LDIM];        // mask[b, s, h]
  __shared__ float wrow[LDIM];        // mrow[s] * (s != h)

  const int tid = threadIdx.x;
  const int blk = blockIdx.x;
  const int bb  = blk >> 7;    // batch
  const int h   = blk & 127;   // head

  // ---------------- init ----------------
  if (tid < LDIM) {
    float mv = mask[(size_t)(bb * LDIM + tid) * LDIM + h] ? 1.f : 0.f;
    mrow[tid] = mv;
    wrow[tid] = (tid == h) ? 0.f : mv;
    const float* pp = s_edge + ((size_t)(bb * LDIM + tid) * LDIM + h) * 2;
    pe[tid][0] = pp[0];
    pe[tid][1] = pp[1];
  }
  for (int i = tid; i < 2 * LDIM * MS; i += 256)
    ((float*)M)[i] = 0.f;
  for (int i = tid; i < LDIM * LDIM; i += 256) {
    int s = i >> 7, d = i & 127;
    float x = s_sib[(((size_t)bb * LDIM + s) * LDIM + h) * LDIM + d];
    C1[s][d] = softplusf(x);
  }
  __syncthreads();

  const int lane = tid & 31;
  const int wave = tid >> 5;
  const int lrow = lane & 15;          // A-matrix row within tile
  const int lkb  = (lane >> 4) << 1;   // K base: 0 (lanes 0-15) / 2 (lanes 16-31)

  for (int it = 0; it < 4; ++it) {
    // ---- phase A: y[v][d] = sum_s wrow[s] * M[v][d][s]  (V_WMMA_F32_16X16X4_F32) ----
    // 16 tile-tasks (2 v-channels x 8 d-tiles), 2 per wave; EXEC all-ones around WMMA.
    for (int task = wave; task < 16; task += 8) {
      const int v  = task >> 3;
      const int d0 = (task & 7) << 4;
      const float* Mr = &M[v][d0 + lrow][0];
      v8f acc = {0.f, 0.f, 0.f, 0.f, 0.f, 0.f, 0.f, 0.f};
      #pragma unroll 4
      for (int k = 0; k < 32; ++k) {
        const int s0 = (k << 2) + lkb;
        v2f afrag, bfrag;
        afrag.x = Mr[s0];
        afrag.y = Mr[s0 + 1];
        bfrag.x = wrow[s0];        // B columns identical -> D columns all equal y[d]
        bfrag.y = wrow[s0 + 1];
        acc = __builtin_amdgcn_wmma_f32_16x16x4_f32(
            false, afrag, false, bfrag, (short)0, acc, false, false);
      }
      // D layout: VGPR r -> row d0+r (lanes 0-15) / d0+8+r (lanes 16-31); take col 0.
      if (lrow == 0) {
        const int rb = d0 + ((lane >> 4) << 3);
        #pragma unroll
        for (int r = 0; r < 8; ++r) yv[v][rb + r] = acc[r];
      }
    }
    __syncthreads();

    // ---- phase B: b[d][v] = pe + mask[d]*(y - w[d]*M[d][d]) ----
    {
      const int d = tid & 127;
      const int v = tid >> 7;
      bv[v][d] = pe[d][v] + mrow[d] * (yv[v][d] - wrow[d] * M[v][d][d]);
    }
    __syncthreads();

    if (it == 3) break;

    // ---- phase C: transposed pairwise message update (in-place, pair-owned) ----
    // Bijective triangular enumeration: 8192 indices, all threads fully active.
    //   d in [0,64), s in [0,128); if s<d reflect to (127-d, 127-s) (strict upper,
    //   row>=64); s==d handles both diagonals (d,d) and (127-d,127-d).
    for (int i = tid; i < 64 * LDIM; i += 256) {
      int d = i >> 7, s = i & 127;
      if (s < d) { d = 127 - d; s = 127 - s; }
      if (s == d) {
        #pragma unroll
        for (int q = 0; q < 2; ++q) {
          const int e = q ? (127 - d) : d;
          const float a0 = bv[0][e] - M[0][e][e] + LN2F;
          const float a1 = bv[1][e] - M[1][e][e] + C1[e][e];
          const float mx = fmaxf(a0, a1);
          const float lse = mx + log1pf(expf(fminf(a0, a1) - mx));
          M[0][e][e] = a0 - lse;
          M[1][e][e] = a1 - lse;
        }
      } else {
        // entry (d,s) reads old M[*][s][d]; mirror (s,d) reads old M[*][d][s]
        const float a0 = bv[0][d] - M[0][s][d] + LN2F;
        const float a1 = bv[1][d] - M[1][s][d] + C1[s][d];
        const float c0 = bv[0][s] - M[0][d][s] + LN2F;
        const float c1 = bv[1][s] - M[1][d][s] + C1[d][s];
        const float mx  = fmaxf(a0, a1);
        const float lse = mx + log1pf(expf(fminf(a0, a1) - mx));
        const float mx2  = fmaxf(c0, c1);
        const float lse2 = mx2 + log1pf(expf(fminf(c0, c1) - mx2));
        M[0][d][s] = a0 - lse;
        M[1][d][s] = a1 - lse;
        M[0][s][d] = c0 - lse2;
        M[1][s][d] = c1 - lse2;
      }
    }
    __syncthreads();
  }

  // ---- output: out[bb, d, h, v] = softmax_v(b[d][v]) ----
  if (tid < LDIM) {
    const int d = tid;
    const float b0 = bv[0][d], b1 = bv[1][d];
    const float mx = fmaxf(b0, b1);
    const float e0 = expf(b0 - mx), e1 = expf(b1 - mx);
    const float inv = 1.f / (e0 + e1);
    float* op = out + (((size_t)bb * LDIM + d) * LDIM + h) * 2;
    op[0] = e0 * inv;
    op[1] = e1 * inv;
  }
}

extern "C" void kernel_launch(void* const* d_in, const int* in_sizes, int n_in,
                              void* d_out, int out_size, void* d_ws, size_t ws_size,
                              hipStream_t stream) {
  (void)in_sizes; (void)n_in; (void)out_size; (void)d_ws; (void)ws_size;
  const float* s_edge = (const float*)d_in[0];
  const float* s_sib  = (const float*)d_in[1];
  const unsigned char* mask = (const unsigned char*)d_in[2];
  float* out = (float*)d_out;
  // one workgroup per (batch, head): 4 * 128 = 512 blocks, 8 waves each
  LoopyBeliefPropagation_72146860638850_kernel<<<dim3(512), dim3(256), 0, stream>>>(
      s_edge, s_sib, mask, out);
}